// SATConv_51505247814287
// MI455X (gfx1250) — compile-verified
//
#include <hip/hip_runtime.h>
#include <stdint.h>

typedef __attribute__((ext_vector_type(2))) float v2f;
typedef __attribute__((ext_vector_type(8))) float v8f;

#define IN_CH 512
#define OUT_CH 256
#define LDS_STRIDE 516  // 512 + 4 pad: stride%64 == 4 -> 16 rows hit 16 distinct banks

// -------------------------------------------------------------------------
// h[n, OUT_CH] = x[n, IN_CH] @ W[IN_CH, OUT_CH], full fp32 via WMMA 16x16x4.
// Block = (32, 16): 16 waves; all waves share one 16-row A tile, staged into
// LDS with GLOBAL_LOAD_ASYNC_TO_LDS_B128 (CDNA5 async path, ASYNCcnt),
// each wave owns one 16-column tile of the output.
// -------------------------------------------------------------------------
__global__ __launch_bounds__(512) void gemm_f32_wmma(
    const float* __restrict__ x, const float* __restrict__ W,
    float* __restrict__ h, int nNodes) {
  __shared__ float ldsA[16 * LDS_STRIDE];

  const int m0 = blockIdx.x * 16;
  const int lane = threadIdx.x;    // 0..31
  const int waveId = threadIdx.y;  // 0..15 -> output column tile
  const int n0 = waveId * 16;
  const int tid = waveId * 32 + lane;  // 0..511

  // ---- async stage of the 16 x 512 A tile into LDS --------------------
  // 2048 16-byte chunks (128 per row); 512 threads x 4 iterations.
  // Chunks are 4-float aligned so they never straddle the padded row stride.
  // VDST = LDS byte address (low 32 bits of the generic shared pointer),
  // VADDR = unsigned byte offset, SADDR = x base (GVS mode).
  const uint32_t ldsBase = (uint32_t)(uintptr_t)(&ldsA[0]);
#pragma unroll
  for (int it = 0; it < 4; ++it) {
    int chunk = tid + it * 512;
    int row = chunk >> 7;          // / 128
    int col = (chunk & 127) << 2;  // 4-float chunks
    int gRow = m0 + row;
    if (gRow >= nNodes) gRow = nNodes - 1;  // clamp; EXEC stays full
    uint32_t lAddr = ldsBase + (uint32_t)(row * LDS_STRIDE + col) * 4u;
    uint32_t gOff = (uint32_t)(gRow * IN_CH + col) * 4u;
    asm volatile("global_load_async_to_lds_b128 %0, %1, %2"
                 :
                 : "v"(lAddr), "v"(gOff), "s"(x)
                 : "memory");
  }
  asm volatile("s_wait_asynccnt 0x0" ::: "memory");
  __syncthreads();

  // A 16x4 f32 layout: lanes 0-15 -> M=lane, VGPR0=K0, VGPR1=K1;
  //                    lanes 16-31 -> M=lane-16, VGPR0=K2, VGPR1=K3.
  // B 4x16 layout mirrors it with N across lanes.
  const int halfSel = lane >> 4;  // 0 or 1
  const int lane15 = lane & 15;
  const int kb = halfSel * 2;
  const float* aRow = &ldsA[lane15 * LDS_STRIDE + kb];
  const float* bCol = &W[(long)kb * OUT_CH + n0 + lane15];

  v8f c = {};
#pragma unroll 8
  for (int k0 = 0; k0 < IN_CH; k0 += 4) {
    v2f a, b;
    a.x = aRow[k0];
    a.y = aRow[k0 + 1];
    b.x = bCol[(long)k0 * OUT_CH];
    b.y = bCol[(long)(k0 + 1) * OUT_CH];
    // 8 args: (neg_a, A, neg_b, B, c_mod, C, reuse_a, reuse_b)
    c = __builtin_amdgcn_wmma_f32_16x16x4_f32(false, a, false, b,
                                              (short)0, c, false, false);
  }

  // C/D layout: lane<16 -> N=lane, M=vgpr i; lane>=16 -> N=lane-16, M=i+8.
  const int mBase = m0 + halfSel * 8;
  const int nIdx = n0 + lane15;
  if (m0 + 16 <= nNodes) {
    // Fast path: full tile, straight-line stores (no exec-mask churn).
    float* hp = &h[(long)mBase * OUT_CH + nIdx];
#pragma unroll
    for (int i = 0; i < 8; ++i) hp[(long)i * OUT_CH] = c[i];
  } else {
#pragma unroll
    for (int i = 0; i < 8; ++i) {
      int row = mBase + i;
      if (row < nNodes) h[(long)row * OUT_CH + nIdx] = c[i];
    }
  }
}

// -------------------------------------------------------------------------
// Degree / normalization / scatter kernels
// -------------------------------------------------------------------------
__global__ void deg_init(float* deg, int n) {
  int i = blockIdx.x * blockDim.x + threadIdx.x;
  if (i < n) deg[i] = 1.0f;  // self loop contributes 1 to every degree
}

__global__ void deg_accum(const long long* __restrict__ ei, float* deg,
                          int nEdges) {
  int e = blockIdx.x * blockDim.x + threadIdx.x;
  if (e < nEdges) {
    long long dst = ei[nEdges + e];  // row 1 of edge_index
    atomicAdd(&deg[dst], 1.0f);
  }
}

__global__ void deg_to_dinv(float* deg, int n) {
  int i = blockIdx.x * blockDim.x + threadIdx.x;
  if (i < n) {
    float d = deg[i];
    deg[i] = d > 0.0f ? rsqrtf(d) : 0.0f;
  }
}

// out[i, c] = dinv[i]^2 * h[i, c]   (self-loop term; also initializes d_out)
__global__ void self_loop_out(const float* __restrict__ h,
                              const float* __restrict__ dinv,
                              float* __restrict__ out, int nNodes) {
  int node = blockIdx.x;
  int c = threadIdx.x;  // 0..255
  if (node < nNodes) {
    float di = dinv[node];
    out[(long)node * OUT_CH + c] = di * di * h[(long)node * OUT_CH + c];
  }
}

#define EDGES_PER_BLOCK 4
// One 256-thread block handles 4 edges; lane = channel. h/out are L2-resident
// (51 MB each << 192 MB L2) so the f32 atomics run at L2 throughput.
__global__ void edge_scatter(const long long* __restrict__ ei,
                             const float* __restrict__ h,
                             const float* __restrict__ dinv,
                             float* __restrict__ out, int nEdges) {
  int c = threadIdx.x;
  int e0 = blockIdx.x * EDGES_PER_BLOCK;
#pragma unroll
  for (int k = 0; k < EDGES_PER_BLOCK; ++k) {
    int e = e0 + k;
    if (e >= nEdges) break;
    long long s = ei[e];
    long long d = ei[nEdges + e];
    float w = dinv[s] * dinv[d];
    atomicAdd(&out[d * (long long)OUT_CH + c],
              w * h[s * (long long)OUT_CH + c]);
  }
}

// -------------------------------------------------------------------------
extern "C" void kernel_launch(void* const* d_in, const int* in_sizes, int n_in,
                              void* d_out, int out_size, void* d_ws,
                              size_t ws_size, hipStream_t stream) {
  const float* x = (const float*)d_in[0];
  const long long* ei = (const long long*)d_in[1];  // int64 per reference
  const float* W = (const float*)d_in[2];
  float* out = (float*)d_out;

  const int nNodes = in_sizes[0] / IN_CH;  // 50000
  const int nEdges = in_sizes[1] / 2;      // 800000

  // Workspace layout: h [nNodes*OUT_CH f32] | deg/dinv [nNodes f32]
  float* h = (float*)d_ws;
  float* deg = h + (size_t)nNodes * OUT_CH;

  // 1) h = x @ W  (fp32 WMMA)
  dim3 gblk(32, 16);
  gemm_f32_wmma<<<(nNodes + 15) / 16, gblk, 0, stream>>>(x, W, h, nNodes);

  // 2) degree with self loops -> dinv (in place)
  deg_init<<<(nNodes + 255) / 256, 256, 0, stream>>>(deg, nNodes);
  deg_accum<<<(nEdges + 255) / 256, 256, 0, stream>>>(ei, deg, nEdges);
  deg_to_dinv<<<(nNodes + 255) / 256, 256, 0, stream>>>(deg, nNodes);

  // 3) out = dinv^2 * h  (self loop, initializes out)
  self_loop_out<<<nNodes, OUT_CH, 0, stream>>>(h, deg, out, nNodes);

  // 4) out[dst] += dinv[src]*dinv[dst] * h[src] over all edges
  edge_scatter<<<(nEdges + EDGES_PER_BLOCK - 1) / EDGES_PER_BLOCK, OUT_CH, 0,
                 stream>>>(ei, h, deg, out, nEdges);
}